// MultiHeadAttention_15092515078512
// MI455X (gfx1250) — compile-verified
//
#include <hip/hip_runtime.h>

// ---------------------------------------------------------------------------
// MultiHeadAttention (no softmax variant) for gfx1250 using fp32 WMMA
//   B=2, S=2048, E=1024, H=16, DH=64
//   q = query @ Wq^T + bq  (head split)   [same for k,v]
//   scores = q k^T / 8 ; masked (mask==0 -> 1e-9) ; out = scores @ v
//   result = merge_heads(out) @ Wo^T + bo
// Data movement: double-buffered GLOBAL_LOAD_ASYNC_TO_LDS_B128 staging
// (ASYNCcnt-pipelined) when available; sync float4 staging otherwise.
// ---------------------------------------------------------------------------

#define E_DIM  1024
#define H_DIM  16
#define DH_DIM 64
#define S_DIM  2048
#define B_DIM  2
#define EPSV   1e-9f

typedef float v2f __attribute__((ext_vector_type(2)));
typedef float v8f __attribute__((ext_vector_type(8)));
typedef int   v4i __attribute__((ext_vector_type(4)));

#if defined(__HIP_DEVICE_COMPILE__)
#if !__has_builtin(__builtin_amdgcn_wmma_f32_16x16x4_f32)
#error "device pass: missing __builtin_amdgcn_wmma_f32_16x16x4_f32"
#endif
#if __has_builtin(__builtin_amdgcn_global_load_async_to_lds_b128)
#define USE_ASYNC_LDS 1
#else
#define USE_ASYNC_LDS 0
#endif
#else
#define USE_ASYNC_LDS 0
#endif

typedef __attribute__((address_space(1))) v4i g_v4i;   // global b128 chunk
typedef __attribute__((address_space(3))) v4i l_v4i;   // LDS b128 chunk

// 16B global -> LDS copy. Async variant is per-lane b128 to LDS, ASYNCcnt.
__device__ __forceinline__ void async_cp16(float* lds_dst, const float* gsrc) {
#if USE_ASYNC_LDS
  __builtin_amdgcn_global_load_async_to_lds_b128(
      (g_v4i*)(float*)gsrc, (l_v4i*)lds_dst, 0, 0);
#else
  *(float4*)lds_dst = *(const float4*)gsrc;
#endif
}

template <int N> __device__ __forceinline__ void wait_async() {
#if USE_ASYNC_LDS
#if __has_builtin(__builtin_amdgcn_s_wait_asynccnt)
  __builtin_amdgcn_s_wait_asynccnt((short)N);
#else
  asm volatile("s_wait_asynccnt %0" ::"n"(N) : "memory");
#endif
#endif
}

// D = A(16x4 f32) * B(4x16 f32) + C(16x16 f32)
__device__ __forceinline__ v8f wmma4(v2f a, v2f b, v8f c) {
  return __builtin_amdgcn_wmma_f32_16x16x4_f32(false, a, false, b,
                                               (short)0, c, false, false);
}

__device__ __forceinline__ v8f zero8() {
  v8f z;
#pragma unroll
  for (int i = 0; i < 8; ++i) z[i] = 0.0f;
  return z;
}

// ---------------------------------------------------------------------------
// C[n0:n0+64, o0:o0+64] = X[n,:] . W[o,:]  (NT gemm, K = E_DIM) + bias[o]
// headSplit: write Out in [B,H,S,DH] layout (o = h*DH + d), else [N, E]
// Block: 128 threads = 4 waves in 2x2; each wave: 2x2 tiles of 16x16.
// Double-buffered LDS staging of 64x16 K-chunks (4 async instr/wave/chunk).
// ---------------------------------------------------------------------------
__global__ __launch_bounds__(128) void gemm_nt(
    const float* __restrict__ X,     // [4096, 1024]
    const float* __restrict__ W,     // [1024, 1024] row-major (out x in)
    const float* __restrict__ bias,  // [1024]
    float* __restrict__ Out,
    int headSplit)
{
  const int n0   = blockIdx.x * 64;
  const int o0   = blockIdx.y * 64;
  const int tid  = threadIdx.x;
  const int wave = tid >> 5;
  const int lane = tid & 31;
  const int wr   = (wave >> 1) * 32;   // wave row offset in 64x64 tile
  const int wc   = (wave & 1) * 32;    // wave col offset
  const int lm   = lane & 15;          // M (A) / N (B,C) index within 16-tile
  const int lh   = lane >> 4;          // K-half select (A/B), +8 rows (C/D)

  // padded stride 20 floats (80B: 16B-aligned columns, b64-aligned frags)
  __shared__ __align__(16) float Xs[2][64 * 20];
  __shared__ __align__(16) float Ws[2][64 * 20];

  // stage one 64x16 chunk of X and W into LDS buffer `bufi`
  auto stage = [&](int e0, int bufi) {
#pragma unroll
    for (int it = 0; it < 2; ++it) {
      int idx = tid + it * 128;          // 0..255
      int r   = idx >> 2;
      int c4  = (idx & 3) * 4;
      async_cp16(&Xs[bufi][r * 20 + c4],
                 X + (size_t)(n0 + r) * E_DIM + e0 + c4);
      async_cp16(&Ws[bufi][r * 20 + c4],
                 W + (size_t)(o0 + r) * E_DIM + e0 + c4);
    }
  };

  v8f acc[2][2];
#pragma unroll
  for (int i = 0; i < 2; ++i)
#pragma unroll
    for (int j = 0; j < 2; ++j) acc[i][j] = zero8();

  stage(0, 0);
  for (int itK = 0; itK < E_DIM / 16; ++itK) {       // 64 iterations
    const int cur = itK & 1;
    if (itK < E_DIM / 16 - 1) {
      stage((itK + 1) * 16, cur ^ 1);
      wait_async<4>();      // retire current buffer, prefetch stays in flight
    } else {
      wait_async<0>();
    }
    __syncthreads();

#pragma unroll
    for (int kk = 0; kk < 4; ++kk) {
      const int co = kk * 4 + lh * 2;      // K column pair within chunk
      v2f a0 = *(const v2f*)(&Xs[cur][(wr + 0  + lm) * 20 + co]);
      v2f a1 = *(const v2f*)(&Xs[cur][(wr + 16 + lm) * 20 + co]);
      v2f b0 = *(const v2f*)(&Ws[cur][(wc + 0  + lm) * 20 + co]);
      v2f b1 = *(const v2f*)(&Ws[cur][(wc + 16 + lm) * 20 + co]);
      acc[0][0] = wmma4(a0, b0, acc[0][0]);
      acc[0][1] = wmma4(a0, b1, acc[0][1]);
      acc[1][0] = wmma4(a1, b0, acc[1][0]);
      acc[1][1] = wmma4(a1, b1, acc[1][1]);
    }
    __syncthreads();   // buffer may be overwritten by next-next stage
  }

  // epilogue: bias + store (C/D layout: vgpr r -> row r + lh*8, col = lm)
#pragma unroll
  for (int tm = 0; tm < 2; ++tm) {
#pragma unroll
    for (int tn = 0; tn < 2; ++tn) {
      const int col = o0 + wc + tn * 16 + lm;
      const float bv = bias[col];
      const int rowBase = n0 + wr + tm * 16 + lh * 8;
#pragma unroll
      for (int r = 0; r < 8; ++r) {
        const int row = rowBase + r;
        const float val = acc[tm][tn][r] + bv;
        size_t off;
        if (headSplit) {
          const int b = row >> 11;         // row / S_DIM
          const int s = row & (S_DIM - 1);
          const int h = col >> 6;          // col / DH
          const int d = col & 63;
          off = (((size_t)(b * H_DIM + h)) * S_DIM + s) * DH_DIM + d;
        } else {
          off = (size_t)row * E_DIM + col;
        }
        Out[off] = val;
      }
    }
  }
}

// ---------------------------------------------------------------------------
// Fused masked-scores attention:  per block: one (b,h), 64 query rows.
//   S = Q K^T * 0.125 ; S = mask ? S : 1e-9 ; O += S V   (loop over k tiles)
// S-GEMM: waves 2x2 over 64x64.  O-GEMM: waves 4x1 (16 q-rows x 64 d-cols).
// K/V tiles double-buffered via async LDS staging (16 async instr/wave/tile).
// ---------------------------------------------------------------------------
__global__ __launch_bounds__(128) void attn(
    const float* __restrict__ Q,   // [B*H, S, DH]
    const float* __restrict__ K,   // [B*H, S, DH]
    const float* __restrict__ V,   // [B*H, S, DH]
    const int*   __restrict__ Msk, // [B, S, S]
    float* __restrict__ Ctx)       // [B, S, E] merged heads
{
  const int q0   = blockIdx.x * 64;
  const int bh   = blockIdx.y;
  const int b    = bh >> 4;
  const int h    = bh & 15;
  const int tid  = threadIdx.x;
  const int wave = tid >> 5;
  const int lane = tid & 31;
  const int wr   = (wave >> 1) * 32;
  const int wc   = (wave & 1) * 32;
  const int lm   = lane & 15;
  const int lh   = lane >> 4;

  // 64x64 tiles, padded stride 68 floats (272B: 16B aligned, b64 OK)
  __shared__ __align__(16) float Ks[2][64 * 68];
  __shared__ __align__(16) float Vs[2][64 * 68];
  __shared__ __align__(16) float Ss[64 * 68];

  const float* Qb = Q + (size_t)bh * S_DIM * DH_DIM;
  const float* Kb = K + (size_t)bh * S_DIM * DH_DIM;
  const float* Vb = V + (size_t)bh * S_DIM * DH_DIM;
  const int*   Mb = Msk + (size_t)b * S_DIM * S_DIM;

  // stage a 64x64 K tile and V tile into buffer `bufi`
  auto stage = [&](int k0, int bufi) {
#pragma unroll
    for (int it = 0; it < 8; ++it) {
      int idx = tid + it * 128;          // 0..1023
      int r   = idx >> 4;
      int c4  = (idx & 15) * 4;
      async_cp16(&Ks[bufi][r * 68 + c4],
                 Kb + (size_t)(k0 + r) * DH_DIM + c4);
      async_cp16(&Vs[bufi][r * 68 + c4],
                 Vb + (size_t)(k0 + r) * DH_DIM + c4);
    }
  };

  stage(0, 0);

  // Q fragments resident in registers: [tm][kk], A-layout (16x4 per frag)
  v2f qf[2][16];
#pragma unroll
  for (int tm = 0; tm < 2; ++tm)
#pragma unroll
    for (int kk = 0; kk < 16; ++kk) {
      const int row = q0 + wr + tm * 16 + lm;
      const int col = kk * 4 + lh * 2;
      qf[tm][kk] = *(const v2f*)(Qb + (size_t)row * DH_DIM + col);
    }

  v8f oacc[4];
#pragma unroll
  for (int tn = 0; tn < 4; ++tn) oacc[tn] = zero8();

  for (int t = 0; t < S_DIM / 64; ++t) {             // 32 k-tiles
    const int cur = t & 1;
    const int k0  = t * 64;
    if (t < S_DIM / 64 - 1) {
      stage(k0 + 64, cur ^ 1);
      wait_async<16>();     // retire current tile; prefetch stays in flight
    } else {
      wait_async<0>();
    }
    __syncthreads();

    // ---- S = Q K^T (NT, K = DH = 64) ----
    v8f sacc[2][2];
#pragma unroll
    for (int i = 0; i < 2; ++i)
#pragma unroll
      for (int j = 0; j < 2; ++j) sacc[i][j] = zero8();

#pragma unroll
    for (int kk = 0; kk < 16; ++kk) {
      const int co = kk * 4 + lh * 2;
      v2f b0 = *(const v2f*)(&Ks[cur][(wc + 0  + lm) * 68 + co]);
      v2f b1 = *(const v2f*)(&Ks[cur][(wc + 16 + lm) * 68 + co]);
      sacc[0][0] = wmma4(qf[0][kk], b0, sacc[0][0]);
      sacc[0][1] = wmma4(qf[0][kk], b1, sacc[0][1]);
      sacc[1][0] = wmma4(qf[1][kk], b0, sacc[1][0]);
      sacc[1][1] = wmma4(qf[1][kk], b1, sacc[1][1]);
    }

    // ---- scale + mask, spill S tile to LDS ----
#pragma unroll
    for (int tm = 0; tm < 2; ++tm) {
#pragma unroll
      for (int tn = 0; tn < 2; ++tn) {
        const int kc = wc + tn * 16 + lm;          // local k column
#pragma unroll
        for (int r = 0; r < 8; ++r) {
          const int qr = wr + tm * 16 + lh * 8 + r; // local q row
          const int mv = Mb[(size_t)(q0 + qr) * S_DIM + (k0 + kc)];
          const float sv = (mv == 0) ? EPSV : sacc[tm][tn][r] * 0.125f;
          Ss[qr * 68 + kc] = sv;
        }
      }
    }
    __syncthreads();

    // ---- O += S V (NN, K = 64); wave owns q-rows [wave*16, wave*16+16) ----
#pragma unroll
    for (int kc = 0; kc < 16; ++kc) {
      const int co = kc * 4 + lh * 2;
      v2f a = *(const v2f*)(&Ss[(wave * 16 + lm) * 68 + co]);
#pragma unroll
      for (int tn = 0; tn < 4; ++tn) {
        v2f bfr;
        bfr.x = Vs[cur][(co + 0) * 68 + tn * 16 + lm];
        bfr.y = Vs[cur][(co + 1) * 68 + tn * 16 + lm];
        oacc[tn] = wmma4(a, bfr, oacc[tn]);
      }
    }
    __syncthreads();   // tile consumed before next-next stage overwrites it
  }

  // ---- write context, merged-head layout [B, S, E] ----
#pragma unroll
  for (int tn = 0; tn < 4; ++tn) {
    const int dcol = tn * 16 + lm;
#pragma unroll
    for (int r = 0; r < 8; ++r) {
      const int qr = wave * 16 + lh * 8 + r;
      Ctx[((size_t)b * S_DIM + (q0 + qr)) * E_DIM + h * DH_DIM + dcol] =
          oacc[tn][r];
    }
  }
}

// ---------------------------------------------------------------------------
extern "C" void kernel_launch(void* const* d_in, const int* in_sizes, int n_in,
                              void* d_out, int out_size, void* d_ws,
                              size_t ws_size, hipStream_t stream) {
  (void)in_sizes; (void)n_in; (void)out_size; (void)ws_size;

  const float* query = (const float*)d_in[0];
  const float* key   = (const float*)d_in[1];
  const float* value = (const float*)d_in[2];
  const int*   mask  = (const int*)  d_in[3];
  const float* Wq    = (const float*)d_in[4];
  const float* bq    = (const float*)d_in[5];
  const float* Wk    = (const float*)d_in[6];
  const float* bk    = (const float*)d_in[7];
  const float* Wv    = (const float*)d_in[8];
  const float* bv    = (const float*)d_in[9];
  const float* Wo    = (const float*)d_in[10];
  const float* bo    = (const float*)d_in[11];

  const size_t matElems = (size_t)B_DIM * H_DIM * S_DIM * DH_DIM; // 4M floats
  float* q_ws = (float*)d_ws;
  float* k_ws = q_ws + matElems;
  float* v_ws = k_ws + matElems;
  float* ctx  = v_ws + matElems;   // needs 4*16MB = 64MB workspace

  dim3 gProj((B_DIM * S_DIM) / 64, E_DIM / 64);   // 64 x 16
  gemm_nt<<<gProj, 128, 0, stream>>>(query, Wq, bq, q_ws, 1);
  gemm_nt<<<gProj, 128, 0, stream>>>(key,   Wk, bk, k_ws, 1);
  gemm_nt<<<gProj, 128, 0, stream>>>(value, Wv, bv, v_ws, 1);

  dim3 gAttn(S_DIM / 64, B_DIM * H_DIM);          // 32 x 32
  attn<<<gAttn, 128, 0, stream>>>(q_ws, k_ws, v_ws, mask, ctx);

  gemm_nt<<<gProj, 128, 0, stream>>>(ctx, Wo, bo, (float*)d_out, 0);
}